// DeepseekV2MoE_31645319037701
// MI455X (gfx1250) — compile-verified
//
#include <hip/hip_runtime.h>
#include <math.h>

typedef __attribute__((ext_vector_type(2))) float v2f;
typedef __attribute__((ext_vector_type(8))) float v8f;
typedef unsigned int v4u __attribute__((ext_vector_type(4)));
typedef int v8i_t __attribute__((ext_vector_type(8)));
typedef int v4i_t __attribute__((ext_vector_type(4)));

// Problem sizes (DeepseekV2 MoE reference)
constexpr int TOK  = 1024;   // tokens
constexpr int HID  = 2048;   // hidden
constexpr int NEXP = 32;     // routed experts
constexpr int FI   = 1408;   // expert intermediate
constexpr int NG   = 8;      // groups
constexpr int KG   = 3;      // top groups
constexpr int KE   = 6;      // top-k experts
constexpr int FSH  = 2816;   // shared intermediate (2*FI)
constexpr float RSCALE = 16.0f;

constexpr int FPAD = FI + 4; // LDS row stride (floats); avoids 64-bank conflicts
constexpr int KC   = 32;     // K-chunk staged per TDM transfer

// --- WMMA f32 16x16x4 wrapper (D = A[16x4] * B[4x16] + C) ---
__device__ __forceinline__ v8f wmma_f32(v2f a, v2f b, v8f c) {
  return __builtin_amdgcn_wmma_f32_16x16x4_f32(false, a, false, b, (short)0, c,
                                               false, false);
}

// LDS byte offset of a __shared__ object: for AMDGPU generic pointers to LDS,
// the low 32 bits are the in-aperture LDS address.
__device__ __forceinline__ unsigned lds_off_of(const void* p) {
  return (unsigned)(uintptr_t)p;
}

// Tensor Data Mover: async 2D tile load Global -> LDS.
// rows = tile_dim1, cols = tile_dim0 (contiguous), row_stride in elements (f32).
// D# built per CDNA5 ISA 8.3/8.4; tracked by TENSORcnt.
// This toolchain's builtin takes 6 args (g0, g1, g2, g3, g4, cpol).
__device__ __forceinline__ void tdm_load_2d(unsigned lds_off, const void* gptr,
                                            int rows, int cols,
                                            int row_stride_elems) {
  const unsigned long long ga = (unsigned long long)(uintptr_t)gptr;
  v4u g0;
  g0[0] = 1u;                                  // count=1 (valid user descriptor)
  g0[1] = lds_off;                             // lds_addr (bytes)
  g0[2] = (unsigned)(ga & 0xffffffffull);      // global_addr[31:0]
  g0[3] = (unsigned)((ga >> 32) & 0x1ffffffull) | (2u << 30); // [56:32] | type=2
  const unsigned dim0 = (unsigned)row_stride_elems;  // tensor_dim0 (>= cols)
  const unsigned dim1 = 0x100000u;                   // tensor_dim1 (large)
  v8i_t g1;
  g1[0] = 0x20000;                                   // data_size = 2 -> 4 bytes
  g1[1] = (int)((dim0 & 0xffffu) << 16);             // tensor_dim0[15:0]
  g1[2] = (int)((dim0 >> 16) | ((dim1 & 0xffffu) << 16));
  g1[3] = (int)((dim1 >> 16) | ((unsigned)cols << 16));   // tile_dim0 = cols
  g1[4] = rows;                                      // tile_dim1 = rows
  g1[5] = (int)dim0;                                 // tensor_dim0_stride lo32
  g1[6] = 0;
  g1[7] = 0;
  v4i_t z4 = {0, 0, 0, 0};                           // groups 2/3 unused (2D)
  v8i_t z8 = {0, 0, 0, 0, 0, 0, 0, 0};
  __builtin_amdgcn_tensor_load_to_lds(g0, g1, z4, z4, z8, 0);
}

// ---------------------------------------------------------------------------
// Kernel 1: router — logits, softmax, grouped top-k -> dense comb[T, E]
// One wave (32 lanes) per token; lane == expert index.
// ---------------------------------------------------------------------------
__global__ __launch_bounds__(256) void moe_router(
    const float* __restrict__ x, const float* __restrict__ wg,
    float* __restrict__ comb) {
  const int wave = threadIdx.x >> 5;
  const int lane = threadIdx.x & 31;
  const int t = blockIdx.x * 8 + wave;

  const float* xr = x + (size_t)t * HID;
  const float* wr = wg + (size_t)lane * HID;
  float acc = 0.f;
  for (int h = 0; h < HID; h += 4) {
    acc += xr[h + 0] * wr[h + 0];
    acc += xr[h + 1] * wr[h + 1];
    acc += xr[h + 2] * wr[h + 2];
    acc += xr[h + 3] * wr[h + 3];
  }
  float m = acc;
  for (int off = 16; off > 0; off >>= 1) m = fmaxf(m, __shfl_xor(m, off, 32));
  float p = __expf(acc - m);
  float s = p;
  for (int off = 16; off > 0; off >>= 1) s += __shfl_xor(s, off, 32);
  const float score = p / s;

  float sc[NEXP];
#pragma unroll
  for (int i = 0; i < NEXP; ++i) sc[i] = __shfl(score, i, 32);

  float gmax[NG];
#pragma unroll
  for (int g = 0; g < NG; ++g) {
    float m4 = sc[4 * g];
    m4 = fmaxf(m4, sc[4 * g + 1]);
    m4 = fmaxf(m4, sc[4 * g + 2]);
    m4 = fmaxf(m4, sc[4 * g + 3]);
    gmax[g] = m4;
  }
  unsigned gsel = 0;
  for (int r = 0; r < KG; ++r) {
    float best = -1.f;
    int bi = 0;
#pragma unroll
    for (int g = 0; g < NG; ++g) {
      const bool avail = ((gsel >> g) & 1u) == 0u;
      if (avail && gmax[g] > best) { best = gmax[g]; bi = g; }
    }
    gsel |= 1u << bi;
  }
  float msc[NEXP];
#pragma unroll
  for (int i = 0; i < NEXP; ++i)
    msc[i] = ((gsel >> (i >> 2)) & 1u) ? sc[i] : 0.f;
  unsigned esel = 0;
  for (int r = 0; r < KE; ++r) {
    float best = -1.f;
    int bi = 0;
#pragma unroll
    for (int i = 0; i < NEXP; ++i) {
      const bool avail = ((esel >> i) & 1u) == 0u;
      if (avail && msc[i] > best) { best = msc[i]; bi = i; }
    }
    esel |= 1u << bi;
  }
  comb[(size_t)t * NEXP + lane] = ((esel >> lane) & 1u) ? msc[lane] : 0.f;
}

// ---------------------------------------------------------------------------
// Kernel 2: deterministic per-expert token compaction (prefix scan, no atomics)
// ---------------------------------------------------------------------------
__global__ __launch_bounds__(1024) void moe_compact(
    const float* __restrict__ comb, int* __restrict__ cnt,
    int* __restrict__ cidx, float* __restrict__ cwgt) {
  __shared__ int sbuf[TOK];
  const int e = blockIdx.x;
  const int t = threadIdx.x;
  const float w = comb[(size_t)t * NEXP + e];
  const int f = (w > 0.f) ? 1 : 0;
  sbuf[t] = f;
  __syncthreads();
  for (int off = 1; off < TOK; off <<= 1) {
    const int v = (t >= off) ? sbuf[t - off] : 0;
    __syncthreads();
    sbuf[t] += v;
    __syncthreads();
  }
  const int incl = sbuf[t];
  if (f) {
    cidx[(size_t)e * TOK + incl - 1] = t;
    cwgt[(size_t)e * TOK + incl - 1] = w * RSCALE;
  }
  if (t == TOK - 1) cnt[e] = incl;
}

// ---------------------------------------------------------------------------
// Kernel 3: shared expert gate_up + SiLU*mul fused  -> act_s[T, FSH]
// ---------------------------------------------------------------------------
__global__ __launch_bounds__(256) void shared_gateup(
    const float* __restrict__ x, const float* __restrict__ ws1,
    float* __restrict__ act_s) {
  const int wave = threadIdx.x >> 5;
  const int lane = threadIdx.x & 31;
  const int hi = lane >> 4, lo = lane & 15;
  const int m0 = blockIdx.x * 16;
  const int n0 = (blockIdx.y * 8 + wave) * 16;

  v8f cg = {}; v8f cu = {};
  for (int k0 = 0; k0 < HID; k0 += 4) {
    v2f a, bg, bu;
    const float* Ar = x + (size_t)(m0 + lo) * HID + k0 + 2 * hi;
    a.x = Ar[0];
    a.y = Ar[1];
    const float* Bg = ws1 + (size_t)(k0 + 2 * hi) * (2 * FSH) + n0 + lo;
    bg.x = Bg[0];
    bg.y = Bg[2 * FSH];
    bu.x = Bg[FSH];
    bu.y = Bg[FSH + 2 * FSH];
    cg = wmma_f32(a, bg, cg);
    cu = wmma_f32(a, bu, cu);
  }
#pragma unroll
  for (int v = 0; v < 8; ++v) {
    const float g = cg[v];
    const float u = cu[v];
    const float act = (g / (1.f + __expf(-g))) * u;  // silu(g)*u
    const int row = m0 + v + 8 * hi;
    act_s[(size_t)row * FSH + n0 + lo] = act;
  }
}

// ---------------------------------------------------------------------------
// Kernel 4: shared expert down proj -> writes (initializes) out[T, H]
// ---------------------------------------------------------------------------
__global__ __launch_bounds__(256) void shared_down(
    const float* __restrict__ act_s, const float* __restrict__ ws2,
    float* __restrict__ out) {
  const int wave = threadIdx.x >> 5;
  const int lane = threadIdx.x & 31;
  const int hi = lane >> 4, lo = lane & 15;
  const int m0 = blockIdx.x * 16;
  const int n0 = (blockIdx.y * 8 + wave) * 16;

  v8f c = {};
  for (int k0 = 0; k0 < FSH; k0 += 4) {
    v2f a, b;
    const float* Ar = act_s + (size_t)(m0 + lo) * FSH + k0 + 2 * hi;
    a.x = Ar[0];
    a.y = Ar[1];
    const float* Br = ws2 + (size_t)(k0 + 2 * hi) * HID + n0 + lo;
    b.x = Br[0];
    b.y = Br[HID];
    c = wmma_f32(a, b, c);
  }
#pragma unroll
  for (int v = 0; v < 8; ++v) {
    const int row = m0 + v + 8 * hi;
    out[(size_t)row * HID + n0 + lo] = c[v];
  }
}

// ---------------------------------------------------------------------------
// Kernel 5: routed experts, fused. TDM (tensor_load_to_lds) double-buffers the
// B tiles into LDS while WMMA consumes the previous chunk from LDS.
// Block = (expert e, 16 compacted tokens). Sparse: exits if tile beyond count.
// ---------------------------------------------------------------------------
__global__ __launch_bounds__(256) void routed_fused(
    const float* __restrict__ x, const float* __restrict__ w1,
    const float* __restrict__ w2, const int* __restrict__ cnt,
    const int* __restrict__ cidx, const float* __restrict__ cwgt,
    float* __restrict__ out) {
  __shared__ float actLDS[16 * FPAD];          // 16 tokens x FI activations
  __shared__ float bstage[8][2][2][KC * 16];   // [wave][buf][gate/up][32x16]
  __shared__ int ridx[16];
  __shared__ float rwgt[16];

  const int e = blockIdx.y;
  const int m0 = blockIdx.x * 16;
  const int count = cnt[e];
  if (m0 >= count) return;  // sparse skip: expert has no tokens in this tile

  if (threadIdx.x < 16) {
    const int r = m0 + threadIdx.x;
    ridx[threadIdx.x] = (r < count) ? cidx[(size_t)e * TOK + r] : 0;
    rwgt[threadIdx.x] = (r < count) ? cwgt[(size_t)e * TOK + r] : 0.f;
  }
  __syncthreads();

  const int wave = threadIdx.x >> 5;
  const int lane = threadIdx.x & 31;
  const int hi = lane >> 4, lo = lane & 15;

  // ---- Phase 1: act[16, FI] into LDS (gathered token rows) ----
  const int rowTok = ridx[lo];
  const float* Arow = x + (size_t)rowTok * HID;
  const float* W1e = w1 + (size_t)e * HID * (2 * FI);
  constexpr int NCH1 = HID / KC;  // 64 K-chunks
  for (int nt = wave; nt < FI / 16; nt += 8) {
    const int n0 = nt * 16;
    const float* Bg0 = W1e + n0;       // gate column panel
    const float* Bu0 = W1e + n0 + FI;  // up column panel
    // prologue: stage chunk 0 into buffer 0 via TDM
    tdm_load_2d(lds_off_of(&bstage[wave][0][0][0]), Bg0, KC, 16, 2 * FI);
    tdm_load_2d(lds_off_of(&bstage[wave][0][1][0]), Bu0, KC, 16, 2 * FI);
    v8f cg = {}; v8f cu = {};
    for (int c = 0; c < NCH1; ++c) {
      if (c + 1 < NCH1) {  // stage next chunk, then wait for current (<=2)
        const int nb = (c + 1) & 1;
        const size_t koff = (size_t)(c + 1) * KC * (2 * FI);
        tdm_load_2d(lds_off_of(&bstage[wave][nb][0][0]), Bg0 + koff, KC, 16,
                    2 * FI);
        tdm_load_2d(lds_off_of(&bstage[wave][nb][1][0]), Bu0 + koff, KC, 16,
                    2 * FI);
        __builtin_amdgcn_s_wait_tensorcnt(2);
      } else {
        __builtin_amdgcn_s_wait_tensorcnt(0);
      }
      const float* Bgl = &bstage[wave][c & 1][0][0];
      const float* Bul = &bstage[wave][c & 1][1][0];
      const int kb = c * KC;
#pragma unroll
      for (int kk = 0; kk < KC; kk += 4) {
        v2f a, bg, bu;
        a.x = Arow[kb + kk + 2 * hi];
        a.y = Arow[kb + kk + 2 * hi + 1];
        bg.x = Bgl[(kk + 2 * hi) * 16 + lo];
        bg.y = Bgl[(kk + 2 * hi + 1) * 16 + lo];
        bu.x = Bul[(kk + 2 * hi) * 16 + lo];
        bu.y = Bul[(kk + 2 * hi + 1) * 16 + lo];
        cg = wmma_f32(a, bg, cg);
        cu = wmma_f32(a, bu, cu);
      }
    }
#pragma unroll
    for (int v = 0; v < 8; ++v) {
      const float g = cg[v];
      const float u = cu[v];
      const float act = (g / (1.f + __expf(-g))) * u;
      const int row = v + 8 * hi;
      actLDS[row * FPAD + n0 + lo] = act;
    }
  }
  __syncthreads();

  // ---- Phase 2: out[tokens, H] += wgt * (act @ w2[e]) ----
  const float* W2e = w2 + (size_t)e * FI * HID;
  constexpr int NCH2 = FI / KC;  // 44 K-chunks
  for (int nt = wave; nt < HID / 16; nt += 8) {
    const int n0 = nt * 16;
    const float* Bb = W2e + n0;
    tdm_load_2d(lds_off_of(&bstage[wave][0][0][0]), Bb, KC, 16, HID);
    v8f cacc = {};
    for (int c = 0; c < NCH2; ++c) {
      if (c + 1 < NCH2) {
        tdm_load_2d(lds_off_of(&bstage[wave][(c + 1) & 1][0][0]),
                    Bb + (size_t)(c + 1) * KC * HID, KC, 16, HID);
        __builtin_amdgcn_s_wait_tensorcnt(1);
      } else {
        __builtin_amdgcn_s_wait_tensorcnt(0);
      }
      const float* Bl = &bstage[wave][c & 1][0][0];
      const int kb = c * KC;
#pragma unroll
      for (int kk = 0; kk < KC; kk += 4) {
        v2f a, b;
        a.x = actLDS[lo * FPAD + kb + kk + 2 * hi];
        a.y = actLDS[lo * FPAD + kb + kk + 2 * hi + 1];
        b.x = Bl[(kk + 2 * hi) * 16 + lo];
        b.y = Bl[(kk + 2 * hi + 1) * 16 + lo];
        cacc = wmma_f32(a, b, cacc);
      }
    }
#pragma unroll
    for (int v = 0; v < 8; ++v) {
      const int row = v + 8 * hi;
      if (m0 + row < count) {
        const float val = cacc[v] * rwgt[row];
        atomicAdd(&out[(size_t)ridx[row] * HID + n0 + lo], val);
      }
    }
  }
}

// ---------------------------------------------------------------------------
extern "C" void kernel_launch(void* const* d_in, const int* in_sizes, int n_in,
                              void* d_out, int out_size, void* d_ws,
                              size_t ws_size, hipStream_t stream) {
  (void)in_sizes; (void)n_in; (void)out_size; (void)ws_size;
  const float* x   = (const float*)d_in[0];  // [T, H]
  const float* wg  = (const float*)d_in[1];  // [E, H]
  const float* w1  = (const float*)d_in[2];  // [E, H, 2F]
  const float* w2  = (const float*)d_in[3];  // [E, F, H]
  const float* ws1 = (const float*)d_in[4];  // [H, 2FS]
  const float* ws2 = (const float*)d_in[5];  // [FS, H]
  float* out = (float*)d_out;                // [T, H]

  // Workspace layout (~11.9 MB total)
  char* p = (char*)d_ws;
  float* comb = (float*)p; p += (size_t)TOK * NEXP * sizeof(float);
  int*   cnt  = (int*)p;   p += (size_t)NEXP * sizeof(int);
  int*   cidx = (int*)p;   p += (size_t)NEXP * TOK * sizeof(int);
  float* cwgt = (float*)p; p += (size_t)NEXP * TOK * sizeof(float);
  float* act_s = (float*)p; /* p += (size_t)TOK * FSH * sizeof(float); */

  moe_router<<<TOK / 8, 256, 0, stream>>>(x, wg, comb);
  moe_compact<<<NEXP, 1024, 0, stream>>>(comb, cnt, cidx, cwgt);
  shared_gateup<<<dim3(TOK / 16, FSH / 128), 256, 0, stream>>>(x, ws1, act_s);
  shared_down<<<dim3(TOK / 16, HID / 128), 256, 0, stream>>>(act_s, ws2, out);
  routed_fused<<<dim3(TOK / 16, NEXP), 256, 0, stream>>>(x, w1, w2, cnt, cidx,
                                                         cwgt, out);
}